// GCN_83511344103765
// MI455X (gfx1250) — compile-verified
//
#include <hip/hip_runtime.h>

// Problem constants (from reference)
#define N_NODES 50000
#define N_EDGES 800000
#define H_DIM   256
#define OUT_DIM 10
#define G_DIM   128

typedef float v2f __attribute__((ext_vector_type(2)));
typedef float v8f __attribute__((ext_vector_type(8)));

// ---------------- utility kernels ----------------

__global__ void k_fill(float* p, long long n, float v) {
    long long i = (long long)blockIdx.x * blockDim.x + threadIdx.x;
    long long stride = (long long)gridDim.x * blockDim.x;
    for (; i < n; i += stride) p[i] = v;
}

__global__ void k_degree(const long long* __restrict__ ei, float* __restrict__ deg) {
    int e = blockIdx.x * blockDim.x + threadIdx.x;
    if (e < N_EDGES) {
        int dst = (int)ei[N_EDGES + e];
        unsafeAtomicAdd(deg + dst, 1.0f);   // deg pre-filled with 1.0 (self-loop)
    }
}

__global__ void k_rsqrt_inplace(float* __restrict__ deg) {
    int i = blockIdx.x * blockDim.x + threadIdx.x;
    if (i < N_NODES) deg[i] = rsqrtf(deg[i]);   // deg >= 1 always (self loop)
}

// ---------------- WMMA f32 GEMM: C[M,256] = A[M,256] * B[256,256] (+bias, relu) ----------
// One wave computes one 16x16 output tile via 64 x V_WMMA_F32_16X16X4_F32.
// Block = 8 waves (256 threads) sharing one 256x16 B column panel staged in LDS.

__global__ void __launch_bounds__(256)
k_gemm_wmma(const float* __restrict__ A, const float* __restrict__ B,
            const float* __restrict__ bias, float* __restrict__ C,
            int M, int fuse_relu) {
    __shared__ float sB[H_DIM * 16];            // B[:, n0:n0+16], layout [k][n], 16 KB

    const int n0 = blockIdx.y * 16;
    {   // cooperative panel load: 256 threads x 16 floats (4x float4 each)
        int r = threadIdx.x;                    // one k-row per thread
        const float4* src = (const float4*)(B + (long long)r * H_DIM + n0);
        float4* dst = (float4*)(sB + r * 16);
        dst[0] = src[0]; dst[1] = src[1]; dst[2] = src[2]; dst[3] = src[3];
    }
    __syncthreads();

    const int wave  = threadIdx.x >> 5;
    const int lane  = threadIdx.x & 31;
    const int mtile = blockIdx.x * 8 + wave;
    if (mtile * 16 >= M) return;                // wave-uniform guard (EXEC all-1 in WMMA)

    const int m0 = mtile * 16;
    const int lo = lane & 15;                   // M index (A) / N index (B,C)
    const int hi = lane >> 4;                   // K sub-select for A/B fragments

    const float* arow = A + (long long)(m0 + lo) * H_DIM;

    v8f c = {0.f, 0.f, 0.f, 0.f, 0.f, 0.f, 0.f, 0.f};

    #pragma unroll 8
    for (int k = 0; k < H_DIM; k += 4) {
        const int kk = k + 2 * hi;
        v2f a;                                  // A[m][kk], A[m][kk+1] (contiguous)
        a.x = arow[kk];
        a.y = arow[kk + 1];
        v2f b;                                  // B[kk][n], B[kk+1][n] from LDS
        b.x = sB[kk * 16 + lo];
        b.y = sB[(kk + 1) * 16 + lo];
        c = __builtin_amdgcn_wmma_f32_16x16x4_f32(
                /*neg_a=*/false, a, /*neg_b=*/false, b,
                /*c_mod=*/(short)0, c, /*reuse_a=*/false, /*reuse_b=*/false);
    }

    const int row_base = m0 + hi * 8;
    const int col = n0 + lo;
    const float bv = bias ? bias[col] : 0.0f;
    #pragma unroll
    for (int r = 0; r < 8; ++r) {
        float v = c[r] + bv;
        if (fuse_relu) v = fmaxf(v, 0.0f);
        C[(long long)(row_base + r) * H_DIM + col] = v;
    }
}

// ---------------- edge scatter: out[dst] += h[src] * norm  (self-loops appended) ----------
// One wave per edge: 8 floats/lane -> coalesced float4 reads, f32 HW atomics out.

__global__ void __launch_bounds__(256)
k_scatter(const float* __restrict__ h, const long long* __restrict__ ei,
          const float* __restrict__ dinv, float* __restrict__ out) {
    const int id = blockIdx.x * 8 + (threadIdx.x >> 5);
    if (id >= N_EDGES + N_NODES) return;
    int src, dst;
    if (id < N_EDGES) { src = (int)ei[id]; dst = (int)ei[N_EDGES + id]; }
    else              { src = id - N_EDGES; dst = src; }
    const float nrm = dinv[src] * dinv[dst];
    const int lane = threadIdx.x & 31;

    const float4* hp = (const float4*)(h + (long long)src * H_DIM + lane * 8);
    float*        op = out + (long long)dst * H_DIM + lane * 8;
    const float4 v0 = hp[0], v1 = hp[1];
    unsafeAtomicAdd(op + 0, v0.x * nrm);
    unsafeAtomicAdd(op + 1, v0.y * nrm);
    unsafeAtomicAdd(op + 2, v0.z * nrm);
    unsafeAtomicAdd(op + 3, v0.w * nrm);
    unsafeAtomicAdd(op + 4, v1.x * nrm);
    unsafeAtomicAdd(op + 5, v1.y * nrm);
    unsafeAtomicAdd(op + 6, v1.z * nrm);
    unsafeAtomicAdd(op + 7, v1.w * nrm);
}

__global__ void k_bias_relu(float* __restrict__ x, const float* __restrict__ b) {
    const long long n = (long long)N_NODES * H_DIM;
    long long i = (long long)blockIdx.x * blockDim.x + threadIdx.x;
    const long long stride = (long long)gridDim.x * blockDim.x;
    for (; i < n; i += stride)
        x[i] = fmaxf(x[i] + b[i & (H_DIM - 1)], 0.0f);
}

// ---------------- global add pool: g[batch[i]] += x[i] ----------------

__global__ void __launch_bounds__(256)
k_pool(const float* __restrict__ x, const long long* __restrict__ batch,
       float* __restrict__ g) {
    const int node = blockIdx.x * 8 + (threadIdx.x >> 5);
    if (node >= N_NODES) return;
    const int lane = threadIdx.x & 31;
    const int gi = (int)batch[node];
    const float4* xp = (const float4*)(x + (long long)node * H_DIM + lane * 8);
    float*        gp = g + (long long)gi * H_DIM + lane * 8;
    const float4 v0 = xp[0], v1 = xp[1];
    unsafeAtomicAdd(gp + 0, v0.x); unsafeAtomicAdd(gp + 1, v0.y);
    unsafeAtomicAdd(gp + 2, v0.z); unsafeAtomicAdd(gp + 3, v0.w);
    unsafeAtomicAdd(gp + 4, v1.x); unsafeAtomicAdd(gp + 5, v1.y);
    unsafeAtomicAdd(gp + 6, v1.z); unsafeAtomicAdd(gp + 7, v1.w);
}

// ---------------- output head: out[g,o] = h1[g,:] . Wm1[:,o] + bm1[o] ----------------

__global__ void k_final(const float* __restrict__ h1, const float* __restrict__ Wm1,
                        const float* __restrict__ bm1, float* __restrict__ out) {
    int i = blockIdx.x * blockDim.x + threadIdx.x;
    if (i >= G_DIM * OUT_DIM) return;
    const int g = i / OUT_DIM, o = i % OUT_DIM;
    const float* hr = h1 + (long long)g * H_DIM;
    float acc = bm1[o];
    #pragma unroll 4
    for (int k = 0; k < H_DIM; ++k) acc += hr[k] * Wm1[k * OUT_DIM + o];
    out[i] = acc;
}

// ---------------- launch ----------------

extern "C" void kernel_launch(void* const* d_in, const int* in_sizes, int n_in,
                              void* d_out, int out_size, void* d_ws, size_t ws_size,
                              hipStream_t stream) {
    (void)in_sizes; (void)n_in; (void)out_size; (void)ws_size;

    const float*      x     = (const float*)d_in[0];
    const long long*  ei    = (const long long*)d_in[1];   // [2, E] int64
    const long long*  batch = (const long long*)d_in[2];   // [N]    int64
    const float* Wc[3] = { (const float*)d_in[3], (const float*)d_in[5], (const float*)d_in[7] };
    const float* bc[3] = { (const float*)d_in[4], (const float*)d_in[6], (const float*)d_in[8] };
    const float* Wm0 = (const float*)d_in[9];
    const float* bm0 = (const float*)d_in[10];
    const float* Wm1 = (const float*)d_in[11];
    const float* bm1 = (const float*)d_in[12];
    float* out = (float*)d_out;

    // workspace layout (floats)
    float* ws   = (float*)d_ws;
    float* deg  = ws;                                     // N       (becomes deg^-1/2)
    float* xbuf = deg  + N_NODES;                         // N*H
    float* hbuf = xbuf + (size_t)N_NODES * H_DIM;         // N*H
    float* gbuf = hbuf + (size_t)N_NODES * H_DIM;         // G*H
    float* h1   = gbuf + (size_t)G_DIM * H_DIM;           // G*H

    // symmetric normalization: deg = 1 (self loop) + indegree; dinv = rsqrt(deg)
    k_fill<<<256, 256, 0, stream>>>(deg, N_NODES, 1.0f);
    k_degree<<<(N_EDGES + 255) / 256, 256, 0, stream>>>(ei, deg);
    k_rsqrt_inplace<<<(N_NODES + 255) / 256, 256, 0, stream>>>(deg);

    const dim3 gemm_grid((N_NODES / 16 + 7) / 8, H_DIM / 16);   // 391 x 16
    const float* cur = x;
    for (int l = 0; l < 3; ++l) {
        k_gemm_wmma<<<gemm_grid, 256, 0, stream>>>(cur, Wc[l], nullptr, hbuf, N_NODES, 0);
        k_fill<<<2048, 256, 0, stream>>>(xbuf, (long long)N_NODES * H_DIM, 0.0f);
        k_scatter<<<(N_EDGES + N_NODES + 7) / 8, 256, 0, stream>>>(hbuf, ei, deg, xbuf);
        k_bias_relu<<<2048, 256, 0, stream>>>(xbuf, bc[l]);
        cur = xbuf;
    }

    // pool to [G, H]
    k_fill<<<64, 256, 0, stream>>>(gbuf, (long long)G_DIM * H_DIM, 0.0f);
    k_pool<<<(N_NODES + 7) / 8, 256, 0, stream>>>(xbuf, batch, gbuf);

    // MLP layer 1 (fused bias+relu, WMMA), then small output head
    const dim3 mlp_grid((G_DIM / 16 + 7) / 8, H_DIM / 16);      // 1 x 16
    k_gemm_wmma<<<mlp_grid, 256, 0, stream>>>(gbuf, Wm0, bm0, h1, G_DIM, 1);
    k_final<<<(G_DIM * OUT_DIM + 255) / 256, 256, 0, stream>>>(h1, Wm1, bm1, out);
}